// FocalLoss_19602230739912
// MI455X (gfx1250) — compile-verified
//
#include <hip/hip_runtime.h>
#include <math.h>

// ---------------------------------------------------------------------------
// Focal attention loss, fused flash-attention style, fp32 WMMA + TDM (gfx1250)
//   sim  = ref_flat(B, N=12288, Fd=256) @ tgt(B, 256, M=4096)
//   p    = softmax over N (online, streamed)
//   pred = rl(B, 8, N) @ p                      (8 accumulators per column)
//   loss = mean((1-pt)^0.5) * (-mean(log_pt))   over B*M pixels, 8 classes
// Ref/target panels are staged global->LDS by the Tensor Data Mover with
// hardware padding (64 data dwords + 8 pad dwords per row = LSTR 72),
// double-buffered so the DMA of tile t+1 overlaps the WMMAs of tile t.
// ---------------------------------------------------------------------------

typedef float v2f __attribute__((ext_vector_type(2)));
typedef float v8f __attribute__((ext_vector_type(8)));
typedef unsigned int u32;
typedef unsigned int v4u __attribute__((ext_vector_type(4)));
typedef int          v8i __attribute__((ext_vector_type(8)));
typedef int          v4i __attribute__((ext_vector_type(4)));

#define B_N    2
#define R_N    3
#define FD     256
#define HW     4096
#define NREF   (R_N * HW)    // 12288
#define M_N    HW            // 4096
#define D_N    8
#define MT     64            // m columns per workgroup (4 waves x 16)
#define NT     64            // n rows per streamed iteration
#define NTILES (NREF / NT)   // 192
#define LSTR   72            // LDS row stride (floats): 64 data + 8 TDM pad

// ---- TDM: 2-D tile (tile_d1 rows x tile_d0 cols) global -> LDS, padded ----
// D# per CDNA5 ISA:
//   g0: count=1 | lds_addr | global_addr | type=2
//   g1: data_size=4B, pad_enable, pad_interval=5 (64 dw), pad_amount=7 (8 dw),
//       tensor_dim0/1, tile_dim0/1, tensor_dim0_stride
__device__ __forceinline__ void tdm_load_tile(u32 lds_byte_addr, const float* gptr,
                                              u32 tensor_d0, u32 tensor_d1,
                                              u32 tile_d0, u32 tile_d1, u32 stride0)
{
    const unsigned long long ga = (unsigned long long)(uintptr_t)gptr;
    v4u g0;
    g0.x = 1u;                                             // count=1, user desc
    g0.y = lds_byte_addr;                                  // LDS dest (bytes)
    g0.z = (u32)(ga & 0xFFFFFFFFu);                        // global addr lo
    g0.w = (u32)((ga >> 32) & 0x01FFFFFFu) | (2u << 30);   // addr hi | type=2
    v8i g1;
    g1[0] = (int)((2u << 16) | (1u << 20) | (5u << 22) | (7u << 25));
    g1[1] = (int)((tensor_d0 & 0xFFFFu) << 16);            // dim0 lo16
    g1[2] = (int)(((tensor_d0 >> 16) & 0xFFFFu) | ((tensor_d1 & 0xFFFFu) << 16));
    g1[3] = (int)(((tensor_d1 >> 16) & 0xFFFFu) | ((tile_d0 & 0xFFFFu) << 16));
    g1[4] = (int)(tile_d1 & 0xFFFFu);                      // tile_dim1, tile_dim2=0
    g1[5] = (int)stride0;                                  // dim0 stride lo32
    g1[6] = 0;
    g1[7] = 0;
    const v4i g2 = (v4i){0, 0, 0, 0};
    const v4i g3 = (v4i){0, 0, 0, 0};
    const v8i g4 = (v8i){0, 0, 0, 0, 0, 0, 0, 0};          // unused groups (2-D tile)
    __builtin_amdgcn_tensor_load_to_lds(g0, g1, g2, g3, g4, 0);
}

__global__ void __launch_bounds__(128)
focal_attn_wmma_kernel(const float* __restrict__ ref,
                       const float* __restrict__ target,
                       const float* __restrict__ ref_label,
                       const int*   __restrict__ target_label,
                       float*       __restrict__ accums)
{
    extern __shared__ float smem[];
    float* tgtS   = smem;                        // [FD][LSTR] target panel
    float* refS0  = smem + FD * LSTR;            // ref panel, buffer 0
    float* refS1  = smem + 2 * FD * LSTR;        // ref panel, buffer 1
    float* rlS    = smem + 3 * FD * LSTR;        // [NT][D_N] labels

    const int b    = blockIdx.y;
    const int m0   = blockIdx.x * MT;
    const int tid  = threadIdx.x;
    const int lane = tid & 31;
    const int wave = tid >> 5;
    const int mcol = lane & 15;                  // column / row within 16-tile
    const int hi   = lane >> 4;                  // half selector (K/M split)

    const u32 ldsBase = (u32)(uintptr_t)(void*)smem;   // LDS aperture: low 32 = offset
    const u32 ldsTgt  = ldsBase;
    const u32 ldsRef[2] = { ldsBase + (u32)(FD * LSTR * 4),
                            ldsBase + (u32)(2 * FD * LSTR * 4) };

    const size_t refBase = (size_t)b * R_N * FD * HW;
    const size_t rlBase  = (size_t)b * R_N * D_N * HW;

    // ---- prologue: TDM the target panel and ref tile 0 (wave 0 issues) ----
    if (wave == 0) {
        tdm_load_tile(ldsTgt, target + (size_t)b * FD * HW + m0,
                      HW, FD, NT, FD, HW);
        tdm_load_tile(ldsRef[0], ref + refBase,      // r=0, hw0=0 for tile 0
                      HW, FD, NT, FD, HW);
    }

    float run_max = -3.0e38f;
    float denom   = 0.0f;
    float acc[D_N];
    #pragma unroll
    for (int d = 0; d < D_N; ++d) acc[d] = 0.0f;

    const int bcol = wave * 16 + mcol;

    for (int t = 0; t < NTILES; ++t) {
        const int p = t & 1;
        // ---- issue TDM for next ref tile into the other buffer ----
        if (wave == 0 && (t + 1) < NTILES) {
            const int n1   = (t + 1) * NT;
            const int r1   = n1 >> 12;
            const int hw1  = n1 & (HW - 1);
            tdm_load_tile(ldsRef[1 - p],
                          ref + refBase + (size_t)r1 * FD * HW + hw1,
                          HW, FD, NT, FD, HW);
        }
        // ---- stage rl panel (tiny, transposed to [nl][d]) ----
        {
            const int n0  = t * NT;
            const int r   = n0 >> 12;
            const int hw0 = n0 & (HW - 1);
            const float* lg = ref_label + rlBase + (size_t)r * D_N * HW + hw0;
            for (int idx = tid; idx < D_N * (NT / 4); idx += 128) {
                const int d = idx >> 4;
                const int v = idx & 15;
                const float4 val = *(const float4*)(lg + (size_t)d * HW + 4 * v);
                rlS[(4 * v + 0) * D_N + d] = val.x;
                rlS[(4 * v + 1) * D_N + d] = val.y;
                rlS[(4 * v + 2) * D_N + d] = val.z;
                rlS[(4 * v + 3) * D_N + d] = val.w;
            }
        }
        // ---- wait: current tile landed (next one may stay in flight) ----
        if (wave == 0) {
            if ((t + 1) < NTILES) __builtin_amdgcn_s_wait_tensorcnt(1);
            else                  __builtin_amdgcn_s_wait_tensorcnt(0);
        }
        __syncthreads();

        const float* refS = p ? refS1 : refS0;

        // ---- sim tile: 64 n-rows x 16 m-cols via V_WMMA_F32_16X16X4_F32 ----
        v8f Dfr[4];
        #pragma unroll
        for (int i = 0; i < 4; ++i) Dfr[i] = (v8f){0,0,0,0,0,0,0,0};

        for (int k = 0; k < FD / 4; ++k) {
            const int fb = 4 * k + 2 * hi;    // K layout: vgpr j + (lane<16?0:2)
            v2f bfrag;
            bfrag.x = tgtS[(fb + 0) * LSTR + bcol];
            bfrag.y = tgtS[(fb + 1) * LSTR + bcol];
            #pragma unroll
            for (int i = 0; i < 4; ++i) {
                v2f afrag;
                const int arow = i * 16 + mcol;   // A row = n within sub-tile
                afrag.x = refS[(fb + 0) * LSTR + arow];
                afrag.y = refS[(fb + 1) * LSTR + arow];
                Dfr[i] = __builtin_amdgcn_wmma_f32_16x16x4_f32(
                    false, afrag, false, bfrag, (short)0, Dfr[i], false, false);
            }
        }

        // ---- online softmax update over this 64-row tile ----
        float tmax = -3.0e38f;
        #pragma unroll
        for (int i = 0; i < 4; ++i)
            #pragma unroll
            for (int v = 0; v < 8; ++v) tmax = fmaxf(tmax, Dfr[i][v]);
        tmax = fmaxf(tmax, __shfl_xor(tmax, 16));   // pair lane holds other 8 rows

        const float nmax = fmaxf(run_max, tmax);
        const float corr = __expf(run_max - nmax);
        run_max = nmax;

        float tsum = 0.0f;
        float tacc[D_N];
        #pragma unroll
        for (int d = 0; d < D_N; ++d) tacc[d] = 0.0f;

        #pragma unroll
        for (int i = 0; i < 4; ++i) {
            #pragma unroll
            for (int v = 0; v < 8; ++v) {
                const float e  = __expf(Dfr[i][v] - nmax);
                tsum += e;
                const int nl   = i * 16 + hi * 8 + v;     // D layout row index
                const float4 rl0 = *(const float4*)(rlS + nl * D_N);
                const float4 rl1 = *(const float4*)(rlS + nl * D_N + 4);
                tacc[0] += e * rl0.x; tacc[1] += e * rl0.y;
                tacc[2] += e * rl0.z; tacc[3] += e * rl0.w;
                tacc[4] += e * rl1.x; tacc[5] += e * rl1.y;
                tacc[6] += e * rl1.z; tacc[7] += e * rl1.w;
            }
        }
        tsum += __shfl_xor(tsum, 16);
        denom = denom * corr + tsum;
        #pragma unroll
        for (int d = 0; d < D_N; ++d) {
            const float tv = tacc[d] + __shfl_xor(tacc[d], 16);
            acc[d] = acc[d] * corr + tv;
        }
        __syncthreads();   // compute done before next iter's TDM / rl staging
    }

    // ---- epilogue: per-column focal terms (each lane owns one column m) ----
    const int   m   = m0 + wave * 16 + mcol;
    const float inv = 1.0f / denom;
    float z[D_N];
    float zmax = -3.0e38f;
    #pragma unroll
    for (int d = 0; d < D_N; ++d) { z[d] = acc[d] * inv; zmax = fmaxf(zmax, z[d]); }
    float se = 0.0f;
    #pragma unroll
    for (int d = 0; d < D_N; ++d) se += __expf(z[d] - zmax);
    const float lse    = __logf(se) + zmax;
    const int   lbl    = target_label[b * HW + m];
    const float log_pt = z[lbl] - lse;
    const float pt     = __expf(log_pt);
    const float focal  = sqrtf(fmaxf(1.0f - pt, 0.0f));

    float v0 = (hi == 0) ? log_pt : 0.0f;     // hi lanes are duplicates
    float v1 = (hi == 0) ? focal  : 0.0f;
    #pragma unroll
    for (int off = 16; off > 0; off >>= 1) {
        v0 += __shfl_xor(v0, off);
        v1 += __shfl_xor(v1, off);
    }
    if (lane == 0) {
        atomicAdd(&accums[0], v0);
        atomicAdd(&accums[1], v1);
    }
}

__global__ void focal_finalize_kernel(const float* __restrict__ accums,
                                      float* __restrict__ out)
{
    const float npix       = (float)(B_N * M_N);
    const float mean_logpt = accums[0] / npix;
    const float mean_focal = accums[1] / npix;
    out[0] = mean_focal * (-mean_logpt);
}

extern "C" void kernel_launch(void* const* d_in, const int* in_sizes, int n_in,
                              void* d_out, int out_size, void* d_ws, size_t ws_size,
                              hipStream_t stream) {
    const float* ref          = (const float*)d_in[0];
    const float* target       = (const float*)d_in[1];
    const float* ref_label    = (const float*)d_in[2];
    const int*   target_label = (const int*)d_in[3];
    float*       out          = (float*)d_out;
    float*       accums       = (float*)d_ws;

    (void)hipMemsetAsync(accums, 0, 2 * sizeof(float), stream);

    const dim3 grid(M_N / MT, B_N);
    const size_t lds_bytes = (size_t)(3 * FD * LSTR + NT * D_N) * sizeof(float);
    focal_attn_wmma_kernel<<<grid, 128, lds_bytes, stream>>>(
        ref, target, ref_label, target_label, accums);
    focal_finalize_kernel<<<1, 1, 0, stream>>>(accums, out);
}